// ts_corr_16149077033489
// MI455X (gfx1250) — compile-verified
//
#include <hip/hip_runtime.h>
#include <hip/hip_bf16.h>

typedef __attribute__((ext_vector_type(2))) float v2f;
typedef __attribute__((ext_vector_type(4))) float v4f;
typedef __attribute__((ext_vector_type(8))) float v8f;

#define NROW 48
#define DWIN 10
#define WSTRIDE 10
#define TLEN 2500
#define NWIN 250
#define NPAIR 1128                 // 48*47/2
#define WCHUNK 16
#define KPAD 12                    // 10 -> 12 (3 WMMA K-steps of 4)
#define CB_WSTR (NROW * KPAD + 4)  // 580: pad so per-window stride != 0 mod 64 banks
#define OB_STR (WCHUNK + 2)        // 18: spread banks for the write-out phase

__global__ __launch_bounds__(256)
void ts_corr_kernel(const float* __restrict__ X, float* __restrict__ out) {
  __shared__ float cbuf[WCHUNK * CB_WSTR];   // centered, K-padded windows
  __shared__ float sbuf[WCHUNK * NROW];      // per-row sigma
  __shared__ float obuf[NPAIR * OB_STR];     // staged corr, [pair][w]
  __shared__ float dump[32];                 // sink for masked-off diag elements

  const int wc  = blockIdx.x;            // window chunk
  const int b   = blockIdx.y;            // batch
  const int tid = threadIdx.x;
  const int w0  = wc * WCHUNK;
  const int wn  = min(WCHUNK, NWIN - w0);   // 16 or 10 (always even)

  // ---- Phase 1: load 10 samples/row, center, sigma -> LDS ----
  {
    const int wl    = tid & 15;        // window within chunk
    const int rbase = tid >> 4;        // 0..15
    if (wl < wn) {
      #pragma unroll
      for (int i = 0; i < 3; ++i) {
        const int row = rbase + i * 16;
        const float* src =
            X + ((size_t)b * NROW + row) * TLEN + (size_t)(w0 + wl) * WSTRIDE;
        float v[DWIN];
        float sum = 0.f;
        #pragma unroll
        for (int k = 0; k < DWIN; ++k) { v[k] = src[k]; sum += v[k]; }
        const float mean = sum * (1.0f / DWIN);
        float* cdst = &cbuf[wl * CB_WSTR + row * KPAD];
        float ss = 0.f;
        #pragma unroll
        for (int k = 0; k < DWIN; ++k) {
          const float c = v[k] - mean;
          ss += c * c;
          cdst[k] = c;
        }
        cdst[10] = 0.f;
        cdst[11] = 0.f;
        sbuf[wl * NROW + row] = sqrtf(ss * (1.0f / (DWIN - 1)));
      }
    }
  }
  __syncthreads();

  // ---- Phase 2: 48x48 Gram via V_WMMA_F32_16X16X4_F32, upper-tri tiles ----
  {
    const int wave = tid >> 5;     // 0..7
    const int lane = tid & 31;
    const int half = lane >> 4;    // 0: lanes 0-15, 1: lanes 16-31
    const int lr   = lane & 15;

    const int tiA[6] = {0, 0, 0, 1, 1, 2};
    const int tjA[6] = {0, 1, 2, 1, 2, 2};

    for (int rep = 0; rep < 2; ++rep) {
      const int wl = wave * 2 + rep;          // wave-uniform
      if (wl >= wn) continue;
      const float* cw = &cbuf[wl * CB_WSTR];
      const float* sw = &sbuf[wl * NROW];

      #pragma unroll
      for (int t6 = 0; t6 < 6; ++t6) {
        const int ti = tiA[t6];
        const int tj = tjA[t6];
        v8f acc = {};
        #pragma unroll
        for (int ks = 0; ks < 3; ++ks) {
          // A 16x4 f32 layout: lanes 0-15 hold K=k0,k0+1; lanes 16-31 K=k0+2,k0+3
          const int k0 = ks * 4 + half * 2;
          const float* ap = &cw[(ti * 16 + lr) * KPAD + k0];
          const float* bp = &cw[(tj * 16 + lr) * KPAD + k0];
          v2f a  = {ap[0], ap[1]};
          v2f bb = {bp[0], bp[1]};
          acc = __builtin_amdgcn_wmma_f32_16x16x4_f32(
              false, a, false, bb, (short)0, acc, false, false);
        }
        // C layout: VGPR r -> (M = r + 8*half, N = lr)
        const int j  = tj * 16 + lr;
        const float sj = sw[j];

        // 8 consecutive sigmas for this lane's rows: 32B-aligned -> 2x b128
        const v4f* sv = (const v4f*)&sw[ti * 16 + half * 8];
        const v4f s_lo = sv[0];
        const v4f s_hi = sv[1];
        float si[8] = {s_lo[0], s_lo[1], s_lo[2], s_lo[3],
                       s_hi[0], s_hi[1], s_hi[2], s_hi[3]};

        if (ti == tj) {
          // diagonal tile: branchless — always store, but route i>=j lanes
          // to a dump slot via address select (v_cndmask, no exec round-trip)
          #pragma unroll
          for (int r = 0; r < 8; ++r) {
            const int i = ti * 16 + r + half * 8;
            const float corr =
                acc[r] * __builtin_amdgcn_rcpf(si[r] * sj + 1e-8f);
            const int p = (i * (95 - i)) / 2 + (j - i - 1);
            float* dst = (i < j) ? &obuf[p * OB_STR + wl] : &dump[lane];
            *dst = corr;
          }
        } else {
          // off-diagonal tile: i < j always holds, no predication
          #pragma unroll
          for (int r = 0; r < 8; ++r) {
            const int i = ti * 16 + r + half * 8;
            const float corr =
                acc[r] * __builtin_amdgcn_rcpf(si[r] * sj + 1e-8f);
            const int p = (i * (95 - i)) / 2 + (j - i - 1);
            obuf[p * OB_STR + wl] = corr;
          }
        }
      }
    }
  }
  __syncthreads();

  // ---- Phase 3: coalesced write-out, one pair-row per thread ----
  // wn is either 16 (full chunk) or 10 (last chunk): specialize both,
  // fully unrolled, so obuf reads merge into ds_load_2addr_b64.
  if (wn == WCHUNK) {
    for (int p = tid; p < NPAIR; p += 256) {
      float* dst = out + ((size_t)b * NPAIR + p) * NWIN + w0;
      const float* srcp = &obuf[p * OB_STR];
      #pragma unroll
      for (int k = 0; k < WCHUNK; k += 2) {   // offsets even -> 8B aligned
        v2f val = {srcp[k], srcp[k + 1]};
        *(v2f*)(dst + k) = val;
      }
    }
  } else {  // wn == 10
    for (int p = tid; p < NPAIR; p += 256) {
      float* dst = out + ((size_t)b * NPAIR + p) * NWIN + w0;
      const float* srcp = &obuf[p * OB_STR];
      #pragma unroll
      for (int k = 0; k < 10; k += 2) {
        v2f val = {srcp[k], srcp[k + 1]};
        *(v2f*)(dst + k) = val;
      }
    }
  }
}

extern "C" void kernel_launch(void* const* d_in, const int* in_sizes, int n_in,
                              void* d_out, int out_size, void* d_ws, size_t ws_size,
                              hipStream_t stream) {
  const float* X = (const float*)d_in[0];
  // d_in[1] = d (10), d_in[2] = stride (10): baked in as compile-time constants.
  float* out = (float*)d_out;

  dim3 grid((NWIN + WCHUNK - 1) / WCHUNK, 128);  // 16 x 128 blocks
  dim3 block(256);
  ts_corr_kernel<<<grid, block, 0, stream>>>(X, out);
}